// ScatterAggregator_49435073577077
// MI455X (gfx1250) — compile-verified
//
#include <hip/hip_runtime.h>

typedef __attribute__((ext_vector_type(2))) float v2f;
typedef __attribute__((ext_vector_type(8))) float v8f;

namespace {
constexpr int  kNPix      = 49152;
constexpr int  kNBuckets  = 8;
constexpr int  kCIn       = 16;
constexpr int  kProjIn    = 136;   // 8*16 + 8
constexpr int  kProjMid   = 128;
constexpr int  kOutDim    = 64;
constexpr int  kRows      = 16 * kNPix;                 // 786432 (nbatch*npix)
constexpr long kGridCells = (long)kRows * kNBuckets;    // 6291456
constexpr float kEps      = 1e-5f;
constexpr int  kWaves     = 4;                          // waves per block
constexpr int  kLdsStride = 132;                        // 128 + pad, even
}

// ---------------------------------------------------------------- zero scratch
__global__ void zero_ws(float4* __restrict__ p, long n4) {
  long i = (long)blockIdx.x * blockDim.x + threadIdx.x;
  long stride = (long)gridDim.x * blockDim.x;
  float4 z; z.x = z.y = z.z = z.w = 0.f;
  for (; i < n4; i += stride) p[i] = z;
}

// ---------------------------------------------------------------- scatter add
__global__ void scatter_add(const float4* __restrict__ x4,
                            const int* __restrict__ bidx,
                            const int* __restrict__ pix,
                            const int* __restrict__ bucket,
                            float* __restrict__ sums,
                            float* __restrict__ counts,
                            int n) {
  int i = blockIdx.x * blockDim.x + threadIdx.x;
  if (i >= n) return;
  long flat = ((long)bidx[i] * kNPix + pix[i]) * kNBuckets + bucket[i];
  float* s = sums + flat * kCIn;
#pragma unroll
  for (int q = 0; q < 4; ++q) {
    float4 v = x4[(long)i * 4 + q];
    __hip_atomic_fetch_add(s + q * 4 + 0, v.x, __ATOMIC_RELAXED, __HIP_MEMORY_SCOPE_AGENT);
    __hip_atomic_fetch_add(s + q * 4 + 1, v.y, __ATOMIC_RELAXED, __HIP_MEMORY_SCOPE_AGENT);
    __hip_atomic_fetch_add(s + q * 4 + 2, v.z, __ATOMIC_RELAXED, __HIP_MEMORY_SCOPE_AGENT);
    __hip_atomic_fetch_add(s + q * 4 + 3, v.w, __ATOMIC_RELAXED, __HIP_MEMORY_SCOPE_AGENT);
  }
  __hip_atomic_fetch_add(counts + flat, 1.0f, __ATOMIC_RELAXED, __HIP_MEMORY_SCOPE_AGENT);
}

// --------------------------------------------- fused GEMM1 + LN + SiLU + GEMM2
__global__ __launch_bounds__(kWaves * 32)
void fused_mlp(const float* __restrict__ sums,
               const float* __restrict__ counts,
               const float* __restrict__ W1, const float* __restrict__ b1,
               const float* __restrict__ gamma, const float* __restrict__ beta,
               const float* __restrict__ W2, const float* __restrict__ b2,
               float* __restrict__ out) {
  __shared__ float hbuf[kWaves * 16 * kLdsStride];
  const int lane = threadIdx.x & 31;
  const int wave = threadIdx.x >> 5;
  const int lo   = lane & 15;
  const int hi   = lane >> 4;
  const int r0   = (blockIdx.x * kWaves + wave) * 16;   // wave's 16 rows
  const int base = wave * 16 * kLdsStride;

  // lane lo owns row r0+lo for A-fragment generation
  const int r = r0 + lo;
  const float* cnt_row = counts + (long)r * kNBuckets;
  const float* sum_row = sums + (long)r * kNBuckets * kCIn;

  // preload this row's 8 bucket stats once; branchless select
  const float4 c0 = *(const float4*)(cnt_row);
  const float4 c1 = *(const float4*)(cnt_row + 4);
  const float carr[8] = {c0.x, c0.y, c0.z, c0.w, c1.x, c1.y, c1.z, c1.w};
  float inv[8], pres[8];
#pragma unroll
  for (int b = 0; b < 8; ++b) {
    const bool p = carr[b] > 0.f;
    inv[b]  = p ? 1.0f / carr[b] : 0.f;
    pres[b] = p ? 1.f : 0.f;
  }

  // ---- GEMM1: h[16x128] = A[16x136] @ W1[136x128]
  v8f acc[8];
#pragma unroll
  for (int t = 0; t < 8; ++t)
#pragma unroll
    for (int j = 0; j < 8; ++j) acc[t][j] = 0.f;

  // K-steps 0..31: aggregated-mean features only (f = 4kt+2hi+{0,1} <= 127).
  // Both k and k+1 stay in bucket kt>>2, so one b64 load + one scale.
#pragma unroll
  for (int kt = 0; kt < 32; ++kt) {
    const int k = kt * 4 + 2 * hi;
    const float2 s2 = *(const float2*)&sum_row[k];
    const float sc = inv[kt >> 2];
    v2f a; a.x = s2.x * sc; a.y = s2.y * sc;
    const float* w1k = W1 + (long)k * kProjMid;
#pragma unroll
    for (int t = 0; t < 8; ++t) {                      // 8 N-tiles of 16
      v2f b;
      b.x = w1k[t * 16 + lo];
      b.y = w1k[kProjMid + t * 16 + lo];
      acc[t] = __builtin_amdgcn_wmma_f32_16x16x4_f32(
          false, a, false, b, (short)0, acc[t], false, false);
    }
  }
  // K-steps 32..33: presence flags only (f >= 128), values already in regs
#pragma unroll
  for (int kt = 32; kt < 34; ++kt) {
    const int fb = (kt - 32) * 4;                      // 0 or 4
    v2f a;
    a.x = hi ? pres[fb + 2] : pres[fb + 0];
    a.y = hi ? pres[fb + 3] : pres[fb + 1];
    const int k = kt * 4 + 2 * hi;
    const float* w1k = W1 + (long)k * kProjMid;
#pragma unroll
    for (int t = 0; t < 8; ++t) {
      v2f b;
      b.x = w1k[t * 16 + lo];
      b.y = w1k[kProjMid + t * 16 + lo];
      acc[t] = __builtin_amdgcn_wmma_f32_16x16x4_f32(
          false, a, false, b, (short)0, acc[t], false, false);
    }
  }

  // ---- +b1, LayerNorm(128), SiLU -> LDS
  float b1v[8], gv[8], bv[8];
#pragma unroll
  for (int t = 0; t < 8; ++t) {
    int n = t * 16 + lo;
    b1v[t] = b1[n]; gv[t] = gamma[n]; bv[t] = beta[n];
  }
  float psum[8], psq[8];
#pragma unroll
  for (int j = 0; j < 8; ++j) { psum[j] = 0.f; psq[j] = 0.f; }
#pragma unroll
  for (int t = 0; t < 8; ++t)
#pragma unroll
    for (int j = 0; j < 8; ++j) {
      float v = acc[t][j] + b1v[t];
      acc[t][j] = v;
      psum[j] += v;
      psq[j]  += v * v;
    }
  // rows 0-7 live in lanes 0-15, rows 8-15 in lanes 16-31 -> reduce within halves
#pragma unroll
  for (int j = 0; j < 8; ++j) {
#pragma unroll
    for (int off = 1; off < 16; off <<= 1) {
      psum[j] += __shfl_xor(psum[j], off, 32);
      psq[j]  += __shfl_xor(psq[j], off, 32);
    }
  }
#pragma unroll
  for (int j = 0; j < 8; ++j) {
    float mu   = psum[j] * (1.f / kProjMid);
    float var  = psq[j] * (1.f / kProjMid) - mu * mu;
    float rstd = rsqrtf(var + kEps);
    int row = j + 8 * hi;
#pragma unroll
    for (int t = 0; t < 8; ++t) {
      float v = (acc[t][j] - mu) * rstd * gv[t] + bv[t];
      v = v * (1.f / (1.f + __expf(-v)));              // SiLU
      hbuf[base + row * kLdsStride + t * 16 + lo] = v;
    }
  }
  asm volatile("s_wait_dscnt 0" ::: "memory");         // wave-local LDS RAW fence

  // ---- GEMM2: out[16x64] = h[16x128] @ W2[128x64]
  v8f acc2[4];
#pragma unroll
  for (int t = 0; t < 4; ++t)
#pragma unroll
    for (int j = 0; j < 8; ++j) acc2[t][j] = 0.f;

#pragma unroll 4
  for (int kt = 0; kt < kProjMid / 4; ++kt) {          // 32 K-steps of 4
    const int k = kt * 4 + 2 * hi;
    const float2 af = *(const float2*)&hbuf[base + lo * kLdsStride + k];
    v2f a; a.x = af.x; a.y = af.y;
    const float* w2k = W2 + (long)k * kOutDim;
#pragma unroll
    for (int t = 0; t < 4; ++t) {                      // 4 N-tiles of 16
      v2f b;
      b.x = w2k[t * 16 + lo];
      b.y = w2k[kOutDim + t * 16 + lo];
      acc2[t] = __builtin_amdgcn_wmma_f32_16x16x4_f32(
          false, a, false, b, (short)0, acc2[t], false, false);
    }
  }

  // ---- +b2, store (element (m,n): lane = (n&15) + 16*(m>=8), vgpr = m&7)
#pragma unroll
  for (int t = 0; t < 4; ++t) {
    float bb = b2[t * 16 + lo];
#pragma unroll
    for (int j = 0; j < 8; ++j) {
      int row = j + 8 * hi;
      out[(long)(r0 + row) * kOutDim + t * 16 + lo] = acc2[t][j] + bb;
    }
  }
}

extern "C" void kernel_launch(void* const* d_in, const int* in_sizes, int n_in,
                              void* d_out, int out_size, void* d_ws, size_t ws_size,
                              hipStream_t stream) {
  // setup_inputs order: x, batch_idx, pix, bucket_id, nbatch, npix,
  //                     W1, b1, gamma, beta, W2, b2
  const float4* x4   = (const float4*)d_in[0];
  const int* bidx    = (const int*)d_in[1];
  const int* pix     = (const int*)d_in[2];
  const int* bucket  = (const int*)d_in[3];
  const float* W1    = (const float*)d_in[6];
  const float* b1    = (const float*)d_in[7];
  const float* gamma = (const float*)d_in[8];
  const float* beta  = (const float*)d_in[9];
  const float* W2    = (const float*)d_in[10];
  const float* b2    = (const float*)d_in[11];
  float* out = (float*)d_out;

  float* sums   = (float*)d_ws;                       // G*16 f32
  float* counts = sums + kGridCells * kCIn;           // G f32

  const int n = in_sizes[1];                          // N points
  const long n4 = kGridCells * (kCIn + 1) / 4;        // f32s/4 to zero

  zero_ws<<<8192, 256, 0, stream>>>((float4*)d_ws, n4);
  scatter_add<<<(n + 255) / 256, 256, 0, stream>>>(x4, bidx, pix, bucket,
                                                   sums, counts, n);
  fused_mlp<<<kRows / (16 * kWaves), kWaves * 32, 0, stream>>>(
      sums, counts, W1, b1, gamma, beta, W2, b2, out);
}